// RGCN_24472723653074
// MI455X (gfx1250) — compile-verified
//
#include <hip/hip_runtime.h>
#include <hip/hip_bf16.h>

#define N_REL 5
#define CH    64

typedef float v2f __attribute__((ext_vector_type(2)));
typedef float v8f __attribute__((ext_vector_type(8)));

// Native fp32 atomic add (GLOBAL_ATOMIC_ADD_F32), never a CAS loop.
__device__ __forceinline__ void fadd_atomic(float* p, float v) {
    unsafeAtomicAdd(p, v);
}

// ---------------- utility: zero fill ----------------
__global__ void zero_kernel(float* __restrict__ p, long long n) {
    long long i = (long long)blockIdx.x * blockDim.x + threadIdx.x;
    long long stride = (long long)gridDim.x * blockDim.x;
    for (; i < n; i += stride) p[i] = 0.0f;
}

// ---------------- per-(node,rel) edge counts ----------------
__global__ void count_kernel(const int* __restrict__ dst,
                             const int* __restrict__ et,
                             float* __restrict__ cnt, int E) {
    int e = blockIdx.x * blockDim.x + threadIdx.x;
    if (e >= E) return;
    fadd_atomic(&cnt[(size_t)dst[e] * N_REL + et[e]], 1.0f);
}

__global__ void inv_kernel(float* __restrict__ cnt, int n) {
    int i = blockIdx.x * blockDim.x + threadIdx.x;
    if (i >= n) return;
    cnt[i] = 1.0f / fmaxf(cnt[i], 1.0f);
}

// ---------------- scatter: agg[dst][rel][c] += feat[src][c] ----------------
// 64 threads (2 waves) per edge. e is wave-uniform: force the index loads
// into SGPRs (scalar/KM path) so the vector pipe only does gather + atomic.
__global__ void scatter_kernel(const float* __restrict__ feat,
                               const int* __restrict__ src,
                               const int* __restrict__ dst,
                               const int* __restrict__ et,
                               float* __restrict__ agg, long long total) {
    long long g = (long long)blockIdx.x * blockDim.x + threadIdx.x;
    if (g >= total) return;
    int e = __builtin_amdgcn_readfirstlane((int)(g >> 6));  // wave-uniform
    int c = (int)(g & 63);
    int s = src[e];
    int d = dst[e];
    int r = et[e];
    fadd_atomic(&agg[(size_t)d * (N_REL * CH) + (size_t)r * CH + c],
                feat[(size_t)s * CH + c]);
}

// ---------------- fused RGCN layer GEMM (WMMA f32 16x16x4) ----------------
// OUT[i] = act( X[i]@ROOT + BIAS + sum_r (AGG[i][r]*INV[i][r]) @ W[r] )
// One wave per 16-row tile; K = 6 segments x 64.
// Layouts per CDNA5 ISA 7.12.2:
//   A (16x4 f32): lane l: M = l&15, half h = l>>4; vgpr0 -> K=2h, vgpr1 -> K=2h+1
//   B (4x16 f32): lane l: N = l&15;               vgpr0 -> K=2h, vgpr1 -> K=2h+1
//   C/D (16x16):  lane l: N = l&15;               vgpr v -> M = 8h + v
template <int DO_RELU>
__global__ __launch_bounds__(256) void rgcn_gemm_kernel(
    const float* __restrict__ X,     // [N,64]
    const float* __restrict__ AGG,   // [N,5,64]
    const float* __restrict__ INV,   // [N,5]
    const float* __restrict__ ROOT,  // [64,64] (K x N row-major)
    const float* __restrict__ W,     // [5,64,64]
    const float* __restrict__ BIAS,  // [64]
    float* __restrict__ OUT,         // [N,64] (may alias X; wave-local rows)
    int n_tiles)
{
    int wave = threadIdx.x >> 5;
    int lane = threadIdx.x & 31;
    int tile = blockIdx.x * 8 + wave;
    if (tile >= n_tiles) return;          // wave-uniform: EXEC stays all-ones
    int row0 = tile * 16;
    int l15  = lane & 15;                 // A: row within tile; B/C: column
    int hh   = lane >> 4;

    v8f acc[4];
#pragma unroll
    for (int j = 0; j < 4; ++j) {
        float bv = BIAS[16 * j + l15];
#pragma unroll
        for (int v = 0; v < 8; ++v) acc[j][v] = bv;
    }

#pragma unroll 1
    for (int seg = 0; seg < 6; ++seg) {
        const float* arow;
        const float* bmat;
        float sc;
        if (seg == 0) {
            arow = X + (size_t)(row0 + l15) * CH;
            bmat = ROOT;
            sc = 1.0f;
        } else {
            int r = seg - 1;
            arow = AGG + (size_t)(row0 + l15) * (N_REL * CH) + (size_t)r * CH;
            bmat = W + (size_t)r * CH * CH;
            sc = INV[(size_t)(row0 + l15) * N_REL + r];
        }
#pragma unroll
        for (int t = 0; t < 16; ++t) {
            int kb = 4 * t + 2 * hh;
            v2f a;
            a.x = arow[kb] * sc;
            a.y = arow[kb + 1] * sc;
#pragma unroll
            for (int j = 0; j < 4; ++j) {
                v2f b;
                b.x = bmat[(size_t)kb * CH + 16 * j + l15];
                b.y = bmat[(size_t)(kb + 1) * CH + 16 * j + l15];
                acc[j] = __builtin_amdgcn_wmma_f32_16x16x4_f32(
                    false, a, false, b, (short)0, acc[j], false, false);
            }
        }
    }

#pragma unroll
    for (int j = 0; j < 4; ++j) {
#pragma unroll
        for (int v = 0; v < 8; ++v) {
            float val = acc[j][v];
            if (DO_RELU) val = fmaxf(val, 0.0f);
            OUT[(size_t)(row0 + 8 * hh + v) * CH + 16 * j + l15] = val;
        }
    }
}

// ---------------- classifier head: logits = h @ Wc + bc ----------------
__global__ void logits_kernel(const float* __restrict__ h,
                              const float* __restrict__ Wc,  // [64,8]
                              const float* __restrict__ bc,  // [8]
                              float* __restrict__ out, int total) {
    int g = blockIdx.x * blockDim.x + threadIdx.x;
    if (g >= total) return;
    int node = g >> 3;
    int c = g & 7;
    float s = bc[c];
    const float* row = h + (size_t)node * CH;
#pragma unroll
    for (int k = 0; k < CH; ++k) s += row[k] * Wc[k * 8 + c];
    out[g] = s;
}

extern "C" void kernel_launch(void* const* d_in, const int* in_sizes, int n_in,
                              void* d_out, int out_size, void* d_ws, size_t ws_size,
                              hipStream_t stream) {
    const float* x     = (const float*)d_in[0];
    const int*   ei    = (const int*)d_in[1];   // [2,E]
    const int*   et    = (const int*)d_in[2];   // [E]
    const float* W1    = (const float*)d_in[3];
    const float* root1 = (const float*)d_in[4];
    const float* b1    = (const float*)d_in[5];
    const float* W2    = (const float*)d_in[6];
    const float* root2 = (const float*)d_in[7];
    const float* b2    = (const float*)d_in[8];
    const float* Wc    = (const float*)d_in[9];
    const float* bc    = (const float*)d_in[10];

    const int N = in_sizes[0] / CH;
    const int E = in_sizes[2];
    const int* src = ei;
    const int* dst = ei + E;

    float* out = (float*)d_out;              // [0,N*64) = h, [N*64,..) = logits
    float* agg = (float*)d_ws;               // N*5*64 floats
    float* inv = agg + (size_t)N * N_REL * CH;  // N*5 floats

    const int n_tiles   = N / 16;            // N = 100000, exact
    const int gemm_grid = (n_tiles + 7) / 8;
    const long long scat_total = (long long)E * CH;
    const int scat_grid = (int)((scat_total + 255) / 256);

    // --- edge counts -> inverse (shared by both layers) ---
    zero_kernel<<<1024, 256, 0, stream>>>(inv, (long long)N * N_REL);
    count_kernel<<<(E + 255) / 256, 256, 0, stream>>>(dst, et, inv, E);
    inv_kernel<<<(N * N_REL + 255) / 256, 256, 0, stream>>>(inv, N * N_REL);

    // --- layer 1: h1 (stored in d_out) = relu(x@root1 + b1 + sum_r mean_r @ W1[r]) ---
    zero_kernel<<<4096, 256, 0, stream>>>(agg, (long long)N * N_REL * CH);
    scatter_kernel<<<scat_grid, 256, 0, stream>>>(x, src, dst, et, agg, scat_total);
    rgcn_gemm_kernel<1><<<gemm_grid, 256, 0, stream>>>(x, agg, inv, root1, W1, b1,
                                                       out, n_tiles);

    // --- layer 2: h (in-place on d_out; each wave only touches its own rows) ---
    zero_kernel<<<4096, 256, 0, stream>>>(agg, (long long)N * N_REL * CH);
    scatter_kernel<<<scat_grid, 256, 0, stream>>>(out, src, dst, et, agg, scat_total);
    rgcn_gemm_kernel<0><<<gemm_grid, 256, 0, stream>>>(out, agg, inv, root2, W2, b2,
                                                       out, n_tiles);

    // --- classifier head ---
    logits_kernel<<<(N * 8 + 255) / 256, 256, 0, stream>>>(
        out, Wc, bc, out + (size_t)N * CH, N * 8);
}